// KNN_Att_8169027797479
// MI455X (gfx1250) — compile-verified
//
#include <hip/hip_runtime.h>
#include <hip/hip_bf16.h>
#include <stdint.h>

typedef float v2f __attribute__((ext_vector_type(2)));
typedef float v4f __attribute__((ext_vector_type(4)));
typedef float v8f __attribute__((ext_vector_type(8)));

#define NROWS    4096
#define IN_FEAT  512
#define OUT_FEAT 256
#define KSEL     32
#define NEGV     (-9.0e15f)

// ---------------------------------------------------------------------------
// helpers
// ---------------------------------------------------------------------------
__device__ __forceinline__ unsigned f2key(float f) {
    unsigned u = __float_as_uint(f);
    return (u & 0x80000000u) ? ~u : (u | 0x80000000u);   // monotonic order-preserving key
}
__device__ __forceinline__ float key2f(unsigned k) {
    return (k & 0x80000000u) ? __uint_as_float(k & 0x7fffffffu) : __uint_as_float(~k);
}

// ---------------------------------------------------------------------------
// K0: zero the norm accumulators
// ---------------------------------------------------------------------------
__global__ void zero_norms(float* __restrict__ n12) {
    int i = blockIdx.x * blockDim.x + threadIdx.x;
    if (i < 2 * NROWS) n12[i] = 0.0f;
}

// ---------------------------------------------------------------------------
// K1: P = M @ W  (M in {X,Y} selected by blockIdx.z) with fp32 WMMA 16x16x4.
//     Block = 256 thr = 8 waves, tile 64(M) x 32(N); wave -> one 16x16 tile.
//     Also accumulates per-row sum of squares into n1/n2 via shuffle+atomic.
// ---------------------------------------------------------------------------
__global__ __launch_bounds__(256) void proj_gemm(
        const float* __restrict__ X, const float* __restrict__ Y,
        const float* __restrict__ W,
        float* __restrict__ Xp, float* __restrict__ Yp,
        float* __restrict__ n1, float* __restrict__ n2) {
    const float* A = (blockIdx.z == 0) ? X : Y;
    float*       P = (blockIdx.z == 0) ? Xp : Yp;
    float*     nrm = (blockIdx.z == 0) ? n1 : n2;

    const int tid  = threadIdx.x;
    const int wave = tid >> 5;
    const int lane = tid & 31;
    const int m0   = blockIdx.x * 64 + (wave & 3) * 16;
    const int n0   = blockIdx.y * 32 + (wave >> 2) * 16;
    const int mr   = lane & 15;           // A: row, B: col (per ISA layout)
    const int ko   = (lane >> 4) * 2;     // lanes 16-31 hold K=2,3

    v8f c = {};
    const float* ap = A + (size_t)(m0 + mr) * IN_FEAT + ko;
    const float* bp = W + (size_t)ko * OUT_FEAT + n0 + mr;
    for (int k = 0; k < IN_FEAT; k += 4) {
        v2f a, b;
        a.x = ap[k];
        a.y = ap[k + 1];
        b.x = bp[(size_t)k * OUT_FEAT];
        b.y = bp[(size_t)(k + 1) * OUT_FEAT];
        c = __builtin_amdgcn_wmma_f32_16x16x4_f32(false, a, false, b,
                                                  (short)0, c, false, false);
    }

#pragma unroll
    for (int r = 0; r < 8; ++r) {
        const int m = m0 + r + 8 * (lane >> 4);
        const int n = n0 + (lane & 15);
        P[(size_t)m * OUT_FEAT + n] = c[r];
        float v = c[r] * c[r];
#pragma unroll
        for (int o = 1; o < 16; o <<= 1) v += __shfl_xor(v, o, 32);
        if ((lane & 15) == 0) atomicAdd(&nrm[m], v);
    }
}

// ---------------------------------------------------------------------------
// K2: norms = sqrt(sumsq)
// ---------------------------------------------------------------------------
__global__ void finalize_norms(float* __restrict__ n12) {
    int i = blockIdx.x * blockDim.x + threadIdx.x;
    if (i < 2 * NROWS) n12[i] = sqrtf(n12[i]);
}

// ---------------------------------------------------------------------------
// K3: cos = (Xp @ Yp^T) / (n1 n2^T + 1e-7); writes cos AND cos^T (both fit L2)
//     Block = 256 thr = 8 waves, tile 64x64; wave -> 16x32 (two accumulators).
//     Tile staged through padded LDS for coalesced transposed stores.
// ---------------------------------------------------------------------------
__global__ __launch_bounds__(256) void cos_gemm(
        const float* __restrict__ Xp, const float* __restrict__ Yp,
        const float* __restrict__ n1, const float* __restrict__ n2,
        float* __restrict__ cosM, float* __restrict__ cosT) {
    __shared__ float tile[64][65];

    const int tid  = threadIdx.x;
    const int wave = tid >> 5;
    const int lane = tid & 31;
    const int m0   = blockIdx.x * 64;
    const int n0   = blockIdx.y * 64;
    const int mr   = lane & 15;
    const int ko   = (lane >> 4) * 2;

    const int wm = (wave & 3) * 16;       // local row of wave tile
    const int wn = (wave >> 2) * 32;      // local col of wave tile

    v8f c0 = {}, c1 = {};
    const float* ar  = Xp + (size_t)(m0 + wm + mr) * OUT_FEAT + ko;
    const float* br0 = Yp + (size_t)(n0 + wn + mr) * OUT_FEAT + ko;       // B = Yp^T
    const float* br1 = Yp + (size_t)(n0 + wn + 16 + mr) * OUT_FEAT + ko;
    for (int k = 0; k < OUT_FEAT; k += 4) {
        v2f a  = *(const v2f*)(ar + k);
        v2f b0 = *(const v2f*)(br0 + k);
        v2f b1 = *(const v2f*)(br1 + k);
        c0 = __builtin_amdgcn_wmma_f32_16x16x4_f32(false, a, false, b0,
                                                   (short)0, c0, false, false);
        c1 = __builtin_amdgcn_wmma_f32_16x16x4_f32(false, a, false, b1,
                                                   (short)0, c1, false, false);
    }

#pragma unroll
    for (int r = 0; r < 8; ++r) {
        const int lm  = wm + r + 8 * (lane >> 4);
        const int lnA = wn + (lane & 15);
        const int lnB = lnA + 16;
        const float nm = n1[m0 + lm];
        tile[lm][lnA] = c0[r] / (nm * n2[n0 + lnA] + 1e-7f);
        tile[lm][lnB] = c1[r] / (nm * n2[n0 + lnB] + 1e-7f);
    }
    __syncthreads();

    // cos (row major), 128-bit stores
    for (int i = tid; i < 64 * 16; i += 256) {
        const int r = i >> 4, cq = (i & 15) << 2;
        v4f v = {tile[r][cq], tile[r][cq + 1], tile[r][cq + 2], tile[r][cq + 3]};
        *(v4f*)(cosM + (size_t)(m0 + r) * NROWS + n0 + cq) = v;
    }
    // cos^T (pad-65 LDS => conflict-free column gathers), 128-bit stores
    for (int i = tid; i < 64 * 16; i += 256) {
        const int r = i >> 4, cq = (i & 15) << 2;   // r = n index, cq = m start
        v4f v = {tile[cq][r], tile[cq + 1][r], tile[cq + 2][r], tile[cq + 3][r]};
        *(v4f*)(cosT + (size_t)(n0 + r) * NROWS + m0 + cq) = v;
    }
}

// ---------------------------------------------------------------------------
// K4: per-row top-k stats. Radix binary search for k-th largest key, then
//     selected-sum reduction.  mode 0: A/S1 (relu sum -> rs = sum^-1/2)
//                              mode 1: B/S2 (sum + (N-k)*NEG -> rs2 = sum^-1)
// ---------------------------------------------------------------------------
__global__ __launch_bounds__(256) void topk_stats(
        const float* __restrict__ M, unsigned* __restrict__ thrOut,
        unsigned* __restrict__ tieOut, float* __restrict__ rsOut, int mode) {
    __shared__ int   shi[8];
    __shared__ float shf[8];
    const int row = blockIdx.x;
    const int tid = threadIdx.x;
    const float* p = M + (size_t)row * NROWS;

    float    vals[16];
    unsigned keys[16];
#pragma unroll
    for (int q = 0; q < 4; ++q) {
        v4f v = *(const v4f*)(p + (size_t)(tid + q * 256) * 4);
#pragma unroll
        for (int j = 0; j < 4; ++j) vals[q * 4 + j] = v[j];
    }
#pragma unroll
    for (int i = 0; i < 16; ++i) keys[i] = f2key(vals[i]);

    unsigned T = 0u;
    for (int b = 31; b >= 0; --b) {
        const unsigned cand = T | (1u << b);
        int cnt = 0;
#pragma unroll
        for (int i = 0; i < 16; ++i) cnt += (keys[i] >= cand) ? 1 : 0;
#pragma unroll
        for (int o = 16; o > 0; o >>= 1) cnt += __shfl_xor(cnt, o, 32);
        if ((tid & 31) == 0) shi[tid >> 5] = cnt;
        __syncthreads();
        int tot = 0;
#pragma unroll
        for (int w = 0; w < 8; ++w) tot += shi[w];
        __syncthreads();
        if (tot >= KSEL) T = cand;
    }

    int   cg = 0;
    float s  = 0.0f;
#pragma unroll
    for (int i = 0; i < 16; ++i) {
        if (keys[i] > T) {
            cg += 1;
            s += (mode == 0) ? fmaxf(vals[i], 0.0f) : vals[i];
        }
    }
#pragma unroll
    for (int o = 16; o > 0; o >>= 1) {
        cg += __shfl_xor(cg, o, 32);
        s  += __shfl_xor(s, o, 32);
    }
    if ((tid & 31) == 0) { shi[tid >> 5] = cg; shf[tid >> 5] = s; }
    __syncthreads();
    if (tid == 0) {
        int cgt = 0; float st = 0.0f;
#pragma unroll
        for (int w = 0; w < 8; ++w) { cgt += shi[w]; st += shf[w]; }
        const int   ties = KSEL - cgt;
        const float tv   = key2f(T);
        float rs;
        if (mode == 0) {
            const float sum = st + (float)ties * fmaxf(tv, 0.0f);
            rs = (sum > 0.0f) ? rsqrtf(sum) : 0.0f;          // inf -> 0 guard
        } else {
            const float sum = st + (float)ties * tv + (float)(NROWS - KSEL) * NEGV;
            const float r   = (sum != 0.0f) ? (1.0f / sum) : 0.0f;
            rs = __builtin_isinf(r) ? 0.0f : r;
        }
        thrOut[row] = T;
        tieOut[row] = (unsigned)ties;
        rsOut[row]  = rs;
    }
}

// ---------------------------------------------------------------------------
// K5: emit output rows.  mode 0: S1[i][j] = sel ? rs_i*rs_j*relu(v) : 0
//                        mode 1: S2[i][j] = rs2_i * (sel ? v : NEG)
//     Non-temporal 128-bit stores: write-once output must not evict cos/cosT
//     from L2.
// ---------------------------------------------------------------------------
__global__ __launch_bounds__(256) void write_out(
        const float* __restrict__ M, const unsigned* __restrict__ thr,
        const unsigned* __restrict__ tie, const float* __restrict__ rs,
        float* __restrict__ out, int mode) {
    __shared__ int tieCnt;
    const int row = blockIdx.x;
    const int tid = threadIdx.x;
    if (tid == 0) tieCnt = 0;
    __syncthreads();

    const unsigned T    = thr[row];
    const int      tb   = (int)tie[row];
    const float    rrow = rs[row];
    const float* p = M + (size_t)row * NROWS;
    float*       o = out + (size_t)row * NROWS;

#pragma unroll
    for (int q = 0; q < 4; ++q) {
        const int base = (tid + q * 256) * 4;
        v4f v = *(const v4f*)(p + base);
        v4f r;
#pragma unroll
        for (int j = 0; j < 4; ++j) {
            const float    vv  = v[j];
            const unsigned key = f2key(vv);
            bool sel = key > T;
            if (!sel && key == T && tb > 0)
                sel = (atomicAdd(&tieCnt, 1) < tb);
            if (mode == 0)
                r[j] = sel ? (fmaxf(vv, 0.0f) * rrow * rs[base + j]) : 0.0f;
            else
                r[j] = rrow * (sel ? vv : NEGV);
        }
        __builtin_nontemporal_store(r, (v4f*)(o + base));
    }
}

// ---------------------------------------------------------------------------
// launcher
// ---------------------------------------------------------------------------
extern "C" void kernel_launch(void* const* d_in, const int* in_sizes, int n_in,
                              void* d_out, int out_size, void* d_ws, size_t ws_size,
                              hipStream_t stream) {
    (void)in_sizes; (void)n_in; (void)out_size;

    const float* X = (const float*)d_in[0];
    const float* Y = (const float*)d_in[1];
    const float* W = (const float*)d_in[2];

    // workspace layout (floats)
    const size_t SZ_P   = (size_t)NROWS * OUT_FEAT;        // 1,048,576
    const size_t SZ_COS = (size_t)NROWS * NROWS;           // 16,777,216
    const size_t need   = (2 * SZ_P + 2 * (size_t)NROWS + 2 * SZ_COS
                           + 6 * (size_t)NROWS) * sizeof(float);
    if (ws_size < need) return;                            // deterministic no-op guard

    float* ws   = (float*)d_ws;
    float* Xp   = ws;
    float* Yp   = Xp + SZ_P;
    float* n1   = Yp + SZ_P;
    float* n2   = n1 + NROWS;
    float* cosM = n2 + NROWS;
    float* cosT = cosM + SZ_COS;
    unsigned* thrA = (unsigned*)(cosT + SZ_COS);
    unsigned* tieA = thrA + NROWS;
    float*    rsA  = (float*)(tieA + NROWS);
    unsigned* thrB = (unsigned*)(rsA + NROWS);
    unsigned* tieB = thrB + NROWS;
    float*    rsB  = (float*)(tieB + NROWS);

    float* S1 = (float*)d_out;
    float* S2 = S1 + SZ_COS;

    zero_norms<<<(2 * NROWS + 255) / 256, 256, 0, stream>>>(n1);

    proj_gemm<<<dim3(NROWS / 64, OUT_FEAT / 32, 2), 256, 0, stream>>>(
        X, Y, W, Xp, Yp, n1, n2);

    finalize_norms<<<(2 * NROWS + 255) / 256, 256, 0, stream>>>(n1);

    cos_gemm<<<dim3(NROWS / 64, NROWS / 64), 256, 0, stream>>>(
        Xp, Yp, n1, n2, cosM, cosT);

    topk_stats<<<NROWS, 256, 0, stream>>>(cosM, thrA, tieA, rsA, 0);
    topk_stats<<<NROWS, 256, 0, stream>>>(cosT, thrB, tieB, rsB, 1);

    write_out<<<NROWS, 256, 0, stream>>>(cosM, thrA, tieA, rsA, S1, 0);
    write_out<<<NROWS, 256, 0, stream>>>(cosT, thrB, tieB, rsB, S2, 1);
}